// LWRHead_19456201851204
// MI455X (gfx1250) — compile-verified
//
#include <hip/hip_runtime.h>

#define Q 1024
#define NTRAIN 20000
#define DIM 128
#define DA 129          // 128 features + intercept (intercept stored at column 128)
#define KSEL 500
#define LSTR 132        // padded row stride (528 B, 16-B aligned) for A / xa staging
#define CH 100          // neighbours staged in LDS per chunk (500 = 5 * 100)
#define RIDGE_F 1e-6f

typedef __attribute__((ext_vector_type(2))) float v2f;
typedef __attribute__((ext_vector_type(8))) float v8f;

// ---------------------------------------------------------------- row norms
__global__ void lwr_norms(const float* __restrict__ M, float* __restrict__ out, int rows) {
  int r = blockIdx.x * blockDim.x + threadIdx.x;
  if (r >= rows) return;
  const float* p = M + (size_t)r * DIM;
  float s = 0.f;
#pragma unroll 8
  for (int j = 0; j < DIM; ++j) s = fmaf(p[j], p[j], s);
  out[r] = s;
}

// ----------------- d2 = |x|^2 - 2 x.t + |t|^2 (fp32 WMMA, 2 N-tiles per wave)
__global__ __launch_bounds__(256) void lwr_dist(const float* __restrict__ Xq,
                                                const float* __restrict__ Xt,
                                                const float* __restrict__ xnorm,
                                                const float* __restrict__ tnorm,
                                                float* __restrict__ d2) {
  int wave = threadIdx.x >> 5;
  int lane = threadIdx.x & 31;
  int half = lane >> 4;      // 0: K 0..1, 1: K 2..3
  int lm   = lane & 15;
  int tile0 = blockIdx.x * 16 + wave * 2;   // two adjacent train tiles per wave
  int tileM = blockIdx.y;                   // query tile (16 rows)
  if (tile0 >= NTRAIN / 16) return;         // wave-uniform exit (1250 is even)

  const float* aptr  = Xq + (size_t)(tileM * 16 + lm) * DIM + half * 2;
  const float* b0ptr = Xt + (size_t)(tile0 * 16 + lm) * DIM + half * 2;
  const float* b1ptr = b0ptr + 16 * DIM;
  v8f c0 = {}, c1 = {};
#pragma unroll 4
  for (int k = 0; k < DIM; k += 4) {
    v2f a, b0, b1;
    a.x  = aptr[k];   a.y  = aptr[k + 1];
    b0.x = b0ptr[k];  b0.y = b0ptr[k + 1];
    b1.x = b1ptr[k];  b1.y = b1ptr[k + 1];
    c0 = __builtin_amdgcn_wmma_f32_16x16x4_f32(false, a, false, b0, (short)0, c0, false, false);
    c1 = __builtin_amdgcn_wmma_f32_16x16x4_f32(false, a, false, b1, (short)0, c1, false, false);
  }
  int gn0 = tile0 * 16 + lm;
  float tn0 = tnorm[gn0];
  float tn1 = tnorm[gn0 + 16];
#pragma unroll
  for (int r = 0; r < 8; ++r) {
    int gq = tileM * 16 + r + 8 * half;     // C layout: VGPR r -> M=r (lo) / 8+r (hi)
    float xn = xnorm[gq];
    d2[(size_t)gq * NTRAIN + gn0]      = fmaxf(xn - 2.0f * c0[r] + tn0, 0.0f);
    d2[(size_t)gq * NTRAIN + gn0 + 16] = fmaxf(xn - 2.0f * c1[r] + tn1, 0.0f);
  }
}

// ------------------------------ per-query exact top-500 via 4-pass radix select
__global__ __launch_bounds__(256) void lwr_select(const float* __restrict__ d2,
                                                  const float* __restrict__ y,
                                                  int* __restrict__ idx_out,
                                                  float* __restrict__ w_out,
                                                  float* __restrict__ wy_out) {
  int q = blockIdx.x;
  int t = threadIdx.x;
  const float* row = d2 + (size_t)q * NTRAIN;
  __shared__ unsigned hist[256];
  __shared__ unsigned prefix_s;
  __shared__ int kneed_s;
  __shared__ int scl[256], sce[256];

  // non-negative fp32: raw bit pattern is order-preserving
  unsigned prefix = 0u;
  int kneed = KSEL;
  for (int pass = 0; pass < 4; ++pass) {
    int shift = 24 - pass * 8;
    unsigned maskHigh = (pass == 0) ? 0u : (0xFFFFFFFFu << (shift + 8));
    hist[t] = 0u;
    __syncthreads();
    for (int i = t; i < NTRAIN; i += 256) {
      unsigned key = __float_as_uint(row[i]);
      if ((key & maskHigh) == prefix)
        atomicAdd(&hist[(key >> shift) & 255u], 1u);
    }
    __syncthreads();
    if (t == 0) {
      int acc2 = 0; int kn = kneed; unsigned bsel = 255u;
      for (int b2 = 0; b2 < 256; ++b2) {
        int c2 = (int)hist[b2];
        if (acc2 + c2 >= kn) { bsel = (unsigned)b2; kn -= acc2; break; }
        acc2 += c2;
      }
      prefix_s = prefix | (bsel << shift);
      kneed_s = kn;
    }
    __syncthreads();
    prefix = prefix_s;
    kneed = kneed_s;
    __syncthreads();
  }
  unsigned T = prefix;   // k-th smallest d2 key

  // deterministic compaction: thread t owns a contiguous segment
  const int seg = (NTRAIN + 255) / 256;   // 79
  int start = t * seg;
  int end = start + seg; if (end > NTRAIN) end = NTRAIN;
  int cl = 0, ce = 0;
  for (int i = start; i < end; ++i) {
    unsigned key = __float_as_uint(row[i]);
    cl += (key < T);
    ce += (key == T);
  }
  scl[t] = cl; sce[t] = ce;
  __syncthreads();
  for (int offd = 1; offd < 256; offd <<= 1) {   // inclusive Hillis-Steele scan
    int a1 = (t >= offd) ? scl[t - offd] : 0;
    int a2 = (t >= offd) ? sce[t - offd] : 0;
    __syncthreads();
    scl[t] += a1; sce[t] += a2;
    __syncthreads();
  }
  int totLess = scl[255];
  int offL = scl[t] - cl;
  int offE = sce[t] - ce;
  int tieBudget = KSEL - totLess;
  float h = sqrtf(__uint_as_float(T)) + 1e-12f;  // bandwidth = k-th distance
  float invh = 1.0f / h;
  int wl = 0, we = 0;
  for (int i = start; i < end; ++i) {
    unsigned key = __float_as_uint(row[i]);
    int pos = -1;
    if (key < T) {
      pos = offL + (wl++);
    } else if (key == T) {
      int tp = offE + (we++);
      if (tp < tieBudget) pos = totLess + tp;
    }
    if (pos >= 0) {
      float dist = sqrtf(__uint_as_float(key));
      float u = fminf(dist * invh, 1.0f);
      float um = 1.0f - u * u * u;
      float wv = um * um * um;                  // tricube
      size_t o = (size_t)q * KSEL + pos;
      idx_out[o] = i;
      w_out[o]  = wv;
      wy_out[o] = wv * y[i];
    }
  }
}

// -------- per-query A = Xa^T diag(w) Xa (+ridge), b = Xa^T (w*y)  (WMMA fp32)
// Neighbour rows staged into LDS with CDNA5 async global->LDS B128 copies.
__global__ __launch_bounds__(288) void lwr_normal(const float* __restrict__ Xt,
                                                  const int* __restrict__ idx,
                                                  const float* __restrict__ w,
                                                  const float* __restrict__ wy,
                                                  float* __restrict__ Aout,
                                                  float* __restrict__ bout) {
  __shared__ __align__(16) float xa[CH * LSTR];  // [CH][0..127]=features, [128]=1.0
  __shared__ float wch[CH];
  __shared__ float wych[CH];
  int q = blockIdx.x;
  int t = threadIdx.x;
  int lane = t & 31, wave = t >> 5;  // 9 waves; wave = A tile-row (i strip)
  int half = lane >> 4, lm = lane & 15;

  v8f acc[9];
  v8f vzero = {};
#pragma unroll
  for (int i = 0; i < 9; ++i) acc[i] = vzero;
  float bacc = 0.f;

  for (int c0 = 0; c0 < KSEL; c0 += CH) {
    __syncthreads();
    for (int r = t; r < CH; r += 288) {
      wch[r]  = w[(size_t)q * KSEL + c0 + r];
      wych[r] = wy[(size_t)q * KSEL + c0 + r];
    }
    // async gather: 100 rows x 512B, 16B per lane-transfer, memory -> LDS direct
    for (int e = t; e < CH * (DIM / 4); e += 288) {
      int r   = e >> 5;            // DIM/4 == 32 transfers per row
      int cc  = (e & 31) * 4;      // feature start (16B aligned both sides)
      int gi  = idx[(size_t)q * KSEL + c0 + r];
      const float* src = Xt + (size_t)gi * DIM + cc;
      unsigned ldsoff = (unsigned)(uintptr_t)(&xa[r * LSTR + cc]); // low 32b = LDS offset
      asm volatile("global_load_async_to_lds_b128 %0, %1, off"
                   :: "v"(ldsoff), "v"(src) : "memory");
    }
    if (t < CH) xa[t * LSTR + DIM] = 1.0f;   // intercept column at index 128
    asm volatile("s_wait_asynccnt 0" ::: "memory");
    __syncthreads();

    int irow = wave * 16 + lm;
    bool iok = irow < DA;
    for (int kk = 0; kk < CH; kk += 4) {
      int kb = kk + half * 2;
      v2f av;                                         // A operand = w_k * Xa[k][i]
      av.x = iok ? wch[kb]     * xa[kb * LSTR + irow]       : 0.f;
      av.y = iok ? wch[kb + 1] * xa[(kb + 1) * LSTR + irow] : 0.f;
#pragma unroll
      for (int tj = 0; tj < 9; ++tj) {                // reuse A across 9 B tiles
        int jcol = tj * 16 + lm;
        bool jok = jcol < DA;
        v2f bv;
        bv.x = jok ? xa[kb * LSTR + jcol]       : 0.f;
        bv.y = jok ? xa[(kb + 1) * LSTR + jcol] : 0.f;
        acc[tj] = __builtin_amdgcn_wmma_f32_16x16x4_f32(false, av, false, bv,
                                                        (short)0, acc[tj], false, false);
      }
    }
    if (t < DA) {
      for (int r = 0; r < CH; ++r) bacc = fmaf(wych[r], xa[r * LSTR + t], bacc);
    }
  }

  float* Aq = Aout + (size_t)q * LSTR * LSTR;
#pragma unroll
  for (int tj = 0; tj < 9; ++tj) {
#pragma unroll
    for (int r = 0; r < 8; ++r) {
      int i = wave * 16 + r + 8 * half;
      int j = tj * 16 + lm;
      if (i < DA && j < DA) {
        float v = acc[tj][r];
        if (i == j) v += RIDGE_F;
        Aq[(size_t)i * LSTR + j] = v;
      }
    }
  }
  if (t < DA) bout[(size_t)q * LSTR + t] = bacc;
}

// --------- per-query GE solve (SPD, no pivoting) + back-sub + predict; 1 wave/query
__global__ __launch_bounds__(32) void lwr_solve(float* __restrict__ Aall,
                                                float* __restrict__ ball,
                                                const float* __restrict__ Xq,
                                                float* __restrict__ preds) {
  int q = blockIdx.x;
  volatile float* A = Aall + (size_t)q * LSTR * LSTR;
  volatile float* b = ball + (size_t)q * LSTR;
  int lane = threadIdx.x;
  __shared__ float prow[DA];
  __shared__ float beta[DA];

  for (int p = 0; p < DA; ++p) {
    for (int j = p + lane; j < DA; j += 32) prow[j] = A[p * LSTR + j];
    float bp = b[p];
    __syncthreads();
    float inv = 1.0f / prow[p];
    for (int r = p + 1 + lane; r < DA; r += 32) {
      float f = A[r * LSTR + p] * inv;
      for (int j = p + 1; j < DA; ++j)
        A[r * LSTR + j] -= f * prow[j];
      b[r] -= f * bp;
    }
    __syncthreads();
  }
  for (int p = DA - 1; p >= 0; --p) {
    float s = (lane == 0) ? b[p] : 0.f;
    for (int j = p + 1 + lane; j < DA; j += 32)
      s -= A[p * LSTR + j] * beta[j];
#pragma unroll
    for (int off = 16; off > 0; off >>= 1) s += __shfl_down(s, off, 32);
    if (lane == 0) beta[p] = s / A[p * LSTR + p];
    __syncthreads();
  }
  // beta[0..127] = feature coefficients, beta[128] = intercept
  float acc = (lane == 0) ? beta[DIM] : 0.f;
  for (int j = lane; j < DIM; j += 32)
    acc = fmaf(Xq[(size_t)q * DIM + j], beta[j], acc);
#pragma unroll
  for (int off = 16; off > 0; off >>= 1) acc += __shfl_down(acc, off, 32);
  if (lane == 0) preds[q] = acc;
}

// ----------------------------------------------------------------- launcher
extern "C" void kernel_launch(void* const* d_in, const int* in_sizes, int n_in,
                              void* d_out, int out_size, void* d_ws, size_t ws_size,
                              hipStream_t stream) {
  (void)in_sizes; (void)n_in; (void)out_size; (void)ws_size;
  const float* X  = (const float*)d_in[0];   // [1024,128]
  const float* Xt = (const float*)d_in[1];   // [20000,128]
  const float* y  = (const float*)d_in[2];   // [20000]
  float* preds = (float*)d_out;              // [1024]

  char* ws = (char*)d_ws;
  size_t off = 0;
  auto take = [&](size_t bytes) -> void* {
    void* p = ws + off;
    off += (bytes + 255) & ~(size_t)255;
    return p;
  };
  float* d2    = (float*)take((size_t)Q * NTRAIN * sizeof(float));       // 80 MB (L2-resident)
  float* xnorm = (float*)take((size_t)Q * sizeof(float));
  float* tnorm = (float*)take((size_t)NTRAIN * sizeof(float));
  int*   idxb  = (int*)  take((size_t)Q * KSEL * sizeof(int));
  float* wbuf  = (float*)take((size_t)Q * KSEL * sizeof(float));
  float* wybuf = (float*)take((size_t)Q * KSEL * sizeof(float));
  float* Amat  = (float*)take((size_t)Q * LSTR * LSTR * sizeof(float));
  float* bvec  = (float*)take((size_t)Q * LSTR * sizeof(float));

  lwr_norms<<<(Q + 255) / 256, 256, 0, stream>>>(X, xnorm, Q);
  lwr_norms<<<(NTRAIN + 255) / 256, 256, 0, stream>>>(Xt, tnorm, NTRAIN);
  lwr_dist<<<dim3((NTRAIN / 16 + 15) / 16, Q / 16), 256, 0, stream>>>(X, Xt, xnorm, tnorm, d2);
  lwr_select<<<Q, 256, 0, stream>>>(d2, y, idxb, wbuf, wybuf);
  lwr_normal<<<Q, 288, 0, stream>>>(Xt, idxb, wbuf, wybuf, Amat, bvec);
  lwr_solve<<<Q, 32, 0, stream>>>(Amat, bvec, X, preds);
}